// Latent_40853728919863
// MI455X (gfx1250) — compile-verified
//
#include <hip/hip_runtime.h>

typedef float v2f __attribute__((ext_vector_type(2)));
typedef float v8f __attribute__((ext_vector_type(8)));

#define NTOT   32768
#define MTOT   2048
#define DDIM   64
#define NSPLIT 16
#define MGRP   16
#define NRANGE (NTOT / NSPLIT)   // 2048 rows per split
#define STEPS  (NRANGE / 16)     // 128 tile steps
#define PITCH  20                // LDS row pitch (floats): 16B-aligned stores, conflict-free reads

__device__ __forceinline__ v8f wmma_f32(v2f a, v2f b, v8f c) {
  // D = A(16x4) * B(4x16) + C(16x16), f32, wave32
  return __builtin_amdgcn_wmma_f32_16x16x4_f32(false, a, false, b, (short)0, c, false, false);
}

// Async copy 16B/lane from global to LDS (GV mode), tracked by ASYNCcnt.
__device__ __forceinline__ void async_copy_b128(uint32_t lds_off, const float* gptr) {
  asm volatile("global_load_async_to_lds_b128 %0, %1, off"
               :: "v"(lds_off), "v"((uint64_t)(uintptr_t)gptr)
               : "memory");
}
__device__ __forceinline__ void wait_async0() {
  asm volatile("s_wait_asynccnt 0x0" ::: "memory");
}

// ---------------------------------------------------------------------------
// Kernel 1: fused  G = zf@e^T  ->  p = exp(alpha*d)  ->  e_numT += P^T @ zf
// Each wave owns a 16-wide m tile; block double-buffers [16n x 64d] zf tiles
// in LDS via global_load_async_to_lds_b128.
// ---------------------------------------------------------------------------
__global__ __launch_bounds__(256) void k_main(const float* __restrict__ z,
                                              const float* __restrict__ e,
                                              const float* __restrict__ logsig,
                                              float* __restrict__ ws_num,
                                              float* __restrict__ ws_sum) {
  __shared__ __align__(16) float T[2][DDIM * PITCH];  // T[buf][d][r] = zf[n0+r, d]

  const int tid  = threadIdx.x;
  const int lane = tid & 31;
  const int wave = tid >> 5;
  const int lm   = lane & 15;
  const int hi   = (lane >> 4) & 1;

  const int mg = blockIdx.x & (MGRP - 1);
  const int ns = blockIdx.x >> 4;
  const int m0 = mg * 128 + wave * 16;

  const float ls    = logsig[0];
  const float alpha = -0.5f * __expf(-2.0f * ls);  // == b in reference
  const float m2a   = -2.0f * alpha;

  // cooperative tile staging: 4 threads per d column, 16B per thread
  const int dcol = tid >> 2;
  const int i4   = (tid & 3) << 2;
  const uint32_t lds_dst[2] = {
      (uint32_t)(uintptr_t)&T[0][dcol * PITCH + i4],
      (uint32_t)(uintptr_t)&T[1][dcol * PITCH + i4]};

  // ---- preload e B-fragments (16 K-chunks) + partial ||e[m]||^2 ----
  v2f   b1[16];
  float emp = 0.f;
  {
    const float* ep = e + (size_t)(m0 + lm) * DDIM + 2 * hi;
    #pragma unroll
    for (int k = 0; k < 16; ++k) {
      v2f b;
      b.x = ep[4 * k + 0];
      b.y = ep[4 * k + 1];
      b1[k] = b;
      emp = fmaf(b.x, b.x, emp);
      emp = fmaf(b.y, b.y, emp);
    }
  }
  const float ae = alpha * (emp + __shfl_xor(emp, 16, 32));  // alpha*||e[m0+lm]||^2

  v8f c2[4];
  #pragma unroll
  for (int t = 0; t < 4; ++t) c2[t] = v8f{};
  float sp = 0.f;

  // issue tile 0 into buffer 0
  {
    const int n0 = ns * NRANGE;
    async_copy_b128(lds_dst[0],
                    z + ((size_t)(n0 >> 10) << 16) + (size_t)dcol * 1024 + (n0 & 1023) + i4);
  }

  for (int s = 0; s < STEPS; ++s) {
    const int cur = s & 1;

    wait_async0();        // this wave's portion of tile s has landed
    __syncthreads();      // all waves' portions landed; prev reads of buf[cur^1] done

    if (s + 1 < STEPS) {  // prefetch tile s+1 into the other buffer
      const int n1 = ns * NRANGE + (s + 1) * 16;
      async_copy_b128(lds_dst[cur ^ 1],
                      z + ((size_t)(n1 >> 10) << 16) + (size_t)dcol * 1024 + (n1 & 1023) + i4);
    }

    const float* Tc = &T[cur][0];

    // ---- GEMM 1: G[n,m] = sum_k zf[n,k]*e[m,k]  (16 wmma, K=64) ----
    v8f   c1  = v8f{};
    float znp = 0.f;
    #pragma unroll
    for (int k = 0; k < 16; ++k) {
      v2f a;
      a.x = Tc[(4 * k + 2 * hi + 0) * PITCH + lm];
      a.y = Tc[(4 * k + 2 * hi + 1) * PITCH + lm];
      znp = fmaf(a.x, a.x, znp);
      znp = fmaf(a.y, a.y, znp);
      c1  = wmma_f32(a, b1[k], c1);
    }
    const float zn = znp + __shfl_xor(znp, 16, 32);  // ||zf[n0+lm]||^2

    // ---- p = exp(alpha*(zn + em - 2G)) in WMMA D-layout ----
    float p[8];
    #pragma unroll
    for (int r = 0; r < 8; ++r) {
      const float znr = __shfl(zn, r + 8 * hi, 32);  // row n0 + r + 8*hi
      p[r] = __expf(fmaf(m2a, c1[r], fmaf(alpha, znr, ae)));
      sp += p[r];
    }

    // ---- GEMM 2: e_newT[m,dd] += P^T @ zf_tile  (16 wmma, K=16) ----
    float sx[8];
    #pragma unroll
    for (int r = 0; r < 8; ++r) sx[r] = __shfl_xor(p[r], 16, 32);

    #pragma unroll
    for (int c = 0; c < 4; ++c) {
      // A2 = P^T rows m, K = n-rows 4c..4c+3  (D-layout -> A-layout via xor16)
      v2f a2;
      if (c == 0) { a2.x = hi ? sx[2] : p[0]; a2.y = hi ? sx[3] : p[1]; }
      if (c == 1) { a2.x = hi ? sx[6] : p[4]; a2.y = hi ? sx[7] : p[5]; }
      if (c == 2) { a2.x = hi ? p[2] : sx[0]; a2.y = hi ? p[3] : sx[1]; }
      if (c == 3) { a2.x = hi ? p[6] : sx[4]; a2.y = hi ? p[7] : sx[5]; }
      #pragma unroll
      for (int t = 0; t < 4; ++t) {
        const float* src = &Tc[(16 * t + lm) * PITCH + 4 * c + 2 * hi];
        v2f b2;
        b2.x = src[0];
        b2.y = src[1];
        c2[t] = wmma_f32(a2, b2, c2[t]);
      }
    }

    __syncthreads();  // reads of buf[cur] done before it is refilled at step s+1
  }

  // ---- write partials for this N-split ----
  const float spt = sp + __shfl_xor(sp, 16, 32);
  if (lane < 16) ws_sum[(size_t)ns * MTOT + m0 + lane] = spt;

  #pragma unroll
  for (int t = 0; t < 4; ++t) {
    #pragma unroll
    for (int r = 0; r < 8; ++r) {
      const int m  = m0 + r + 8 * hi;
      const int dd = 16 * t + lm;
      ws_num[((size_t)ns * MTOT + m) * DDIM + dd] = c2[t][r];
    }
  }
}

// ---------------------------------------------------------------------------
// Kernel 2: combine N-splits, normalize e_new, per-m lse
// ---------------------------------------------------------------------------
__global__ __launch_bounds__(64) void k_reduce(const float* __restrict__ ws_num,
                                               const float* __restrict__ ws_sum,
                                               float* __restrict__ ws_lse,
                                               float* __restrict__ out_e) {
  const int m  = blockIdx.x;
  const int dd = threadIdx.x;
  float acc = 0.f, ssum = 0.f;
  #pragma unroll
  for (int ns = 0; ns < NSPLIT; ++ns) {
    acc  += ws_num[((size_t)ns * MTOT + m) * DDIM + dd];
    ssum += ws_sum[(size_t)ns * MTOT + m];
  }
  out_e[(size_t)dd * MTOT + m] = acc / ssum;   // e_new[D][M]
  if (dd == 0) ws_lse[m] = logf(ssum);         // lse = log sum exp(alpha*d)
}

// ---------------------------------------------------------------------------
// Kernel 3: loss = -mean(lse) + 0.5*D*(2*ls - 1) + log(N)
// ---------------------------------------------------------------------------
__global__ __launch_bounds__(256) void k_loss(const float* __restrict__ ws_lse,
                                              const float* __restrict__ logsig,
                                              float* __restrict__ out_loss) {
  __shared__ float red[8];
  float s = 0.f;
  for (int i = threadIdx.x; i < MTOT; i += 256) s += ws_lse[i];
  #pragma unroll
  for (int off = 16; off > 0; off >>= 1) s += __shfl_xor(s, off, 32);
  if ((threadIdx.x & 31) == 0) red[threadIdx.x >> 5] = s;
  __syncthreads();
  if (threadIdx.x == 0) {
    float tot = 0.f;
    #pragma unroll
    for (int w = 0; w < 8; ++w) tot += red[w];
    const float ls = logsig[0];
    out_loss[0] = -(tot / (float)MTOT)
                  + 0.5f * (float)DDIM * (2.f * ls - 1.f)
                  + logf((float)NTOT);
  }
}

extern "C" void kernel_launch(void* const* d_in, const int* in_sizes, int n_in,
                              void* d_out, int out_size, void* d_ws, size_t ws_size,
                              hipStream_t stream) {
  (void)in_sizes; (void)n_in; (void)out_size; (void)ws_size;
  const float* z      = (const float*)d_in[0];
  const float* e      = (const float*)d_in[1];
  const float* logsig = (const float*)d_in[2];

  float* ws_num = (float*)d_ws;                                   // 16*2048*64
  float* ws_sum = ws_num + (size_t)NSPLIT * MTOT * DDIM;          // 16*2048
  float* ws_lse = ws_sum + (size_t)NSPLIT * MTOT;                 // 2048

  float* out      = (float*)d_out;
  float* out_loss = out;
  float* out_e    = out + 1;

  k_main  <<<NSPLIT * MGRP, 256, 0, stream>>>(z, e, logsig, ws_num, ws_sum);
  k_reduce<<<MTOT,          64,  0, stream>>>(ws_num, ws_sum, ws_lse, out_e);
  k_loss  <<<1,             256, 0, stream>>>(ws_lse, logsig, out_loss);
}